// SAGEMeanConv_86492051407618
// MI455X (gfx1250) — compile-verified
//
#include <hip/hip_runtime.h>
#include <hip/hip_bf16.h>

typedef __attribute__((ext_vector_type(2))) float v2f;
typedef __attribute__((ext_vector_type(4))) float v4f;
typedef __attribute__((ext_vector_type(8))) float v8f;

#define KDIM 128
#define NDIM 128
#define WAVES_PER_BLOCK 8

// Fused: out = l2_normalize_rows( X @ W + bias )
// (scatter_mean(x[row], row) == x algebraically, so aggregation is elided)
//
// LDS W layout is pair-interleaved: Wp[p * NDIM + c] = (W[2p][c], W[2p+1][c]),
// so a B fragment for WMMA f32 16x16x4 (lane needs rows {k0+2h, k0+2h+1} at
// its column) is ONE aligned ds_load_b64 straight into the operand VGPR pair
// (pairs of them combine into ds_load_2addr_b64).
__global__ __launch_bounds__(256) void sage_fused_gemm_norm(
    const float* __restrict__ x,     // [M, 128]
    const float* __restrict__ w,     // [128, 128] row-major (in_c, out_c)
    const float* __restrict__ bias,  // [128]
    float* __restrict__ out,         // [M, 128]
    int M) {
  __shared__ v2f Wlds[(KDIM / 2) * NDIM];  // 64 KB, pair-interleaved

  // Cooperative interleaved fill: thread t owns row r = t/2, 64 cols starting
  // at (t&1)*64. One-time 64 KB transpose-pack; cost is negligible.
  {
    const int tid = threadIdx.x;
    const int r = tid >> 1;
    const int c0 = (tid & 1) << 6;
    float* Wf = (float*)Wlds;
    const int pr = (r >> 1) * NDIM;  // pair-row base (in v2f units)
    const int par = r & 1;           // which half of the pair
#pragma unroll
    for (int i = 0; i < 64; i += 4) {
      v4f v = *(const v4f*)(w + r * NDIM + c0 + i);
      Wf[(pr + c0 + i + 0) * 2 + par] = v.x;
      Wf[(pr + c0 + i + 1) * 2 + par] = v.y;
      Wf[(pr + c0 + i + 2) * 2 + par] = v.z;
      Wf[(pr + c0 + i + 3) * 2 + par] = v.w;
    }
  }
  __syncthreads();

  const int lane = threadIdx.x & 31;
  const int wv   = threadIdx.x >> 5;
  const int half = lane >> 4;   // 0: lanes 0-15, 1: lanes 16-31
  const int ln   = lane & 15;

  int base_m = (blockIdx.x * WAVES_PER_BLOCK + wv) * 16;
  if (base_m >= M) return;  // wave-uniform; no barriers past this point

  // A-fragment source row (clamped so EXEC stays all-ones for WMMA).
  int row = base_m + ln;
  if (row > M - 1) row = M - 1;
  const float* xrow = x + (size_t)row * KDIM + half * 2;
  const v2f* wbase = &Wlds[half * NDIM + ln];

  v8f acc[8] = {};  // 8 N-tiles of 16x16 f32 C/D

  // Fully unrolled K loop (32 steps of K=4): straight-line code, no loop
  // backedge, so no accumulator rotation copies; scheduler hoists the 32
  // global_load_b64 A-fragment loads ahead of their WMMA groups.
#pragma unroll
  for (int k0 = 0; k0 < KDIM; k0 += 4) {
    v2f a = *(const v2f*)(xrow + k0);
    const v2f* wl = wbase + (k0 >> 1) * NDIM;
#pragma unroll
    for (int t = 0; t < 8; ++t) {
      v2f b = wl[t * 16];  // ds b64: (W[k0+2h][col], W[k0+2h+1][col])
      acc[t] = __builtin_amdgcn_wmma_f32_16x16x4_f32(
          false, a, false, b, (short)0, acc[t], false, false);
    }
  }

  // Bias: each acc VGPR component is one output element at column 16t+ln.
#pragma unroll
  for (int t = 0; t < 8; ++t) {
    float bv = bias[t * 16 + ln];
#pragma unroll
    for (int g = 0; g < 8; ++g) acc[t][g] += bv;
  }

  // Row L2 norms. C/D layout: VGPR g holds row (g + 8*half) at col ln per
  // half-wave, so a 16-lane butterfly gives the full 128-col sum of squares.
  float inv[8];
#pragma unroll
  for (int g = 0; g < 8; ++g) {
    float s = 0.f;
#pragma unroll
    for (int t = 0; t < 8; ++t) s += acc[t][g] * acc[t][g];
    s += __shfl_xor(s, 1, 32);
    s += __shfl_xor(s, 2, 32);
    s += __shfl_xor(s, 4, 32);
    s += __shfl_xor(s, 8, 32);
    float n = sqrtf(s);
    inv[g] = 1.0f / fmaxf(n, 1e-12f);  // matches out / max(norm, eps)
  }

  // Store: each 16-lane half writes a contiguous 64B segment per (g, t).
#pragma unroll
  for (int g = 0; g < 8; ++g) {
    int m = base_m + g + 8 * half;
    if (m < M) {
      float* orow = out + (size_t)m * NDIM + ln;
#pragma unroll
      for (int t = 0; t < 8; ++t) orow[t * 16] = acc[t][g] * inv[g];
    }
  }
}

extern "C" void kernel_launch(void* const* d_in, const int* in_sizes, int n_in,
                              void* d_out, int out_size, void* d_ws, size_t ws_size,
                              hipStream_t stream) {
  const float* x    = (const float*)d_in[0];
  // d_in[1] = edge_index (int64) — unused: scatter_mean(x[row], row) == x.
  const float* w    = (const float*)d_in[2];
  const float* bias = (const float*)d_in[3];
  float* out        = (float*)d_out;

  const int M = in_sizes[0] / KDIM;  // 50000
  const int wave_tiles = (M + 15) / 16;
  const int blocks = (wave_tiles + WAVES_PER_BLOCK - 1) / WAVES_PER_BLOCK;

  sage_fused_gemm_norm<<<blocks, 256, 0, stream>>>(x, w, bias, out, M);
}